// MtpDrafterModel_12506944766514
// MI455X (gfx1250) — compile-verified
//
#include <hip/hip_runtime.h>
#include <hip/hip_bf16.h>

typedef __attribute__((ext_vector_type(2))) float v2f;
typedef __attribute__((ext_vector_type(8))) float v8f;

#define EPSN 1e-6f

// ---------------------------------------------------------------------------
// Generic small-M GEMM via V_WMMA_F32_16X16X4_F32.
// out(Mrows x N) = A(Mrows x K, ld=K) @ W(K x N, row major), Mrows == 8.
// One wave computes one 16x16 output tile. Lanes mapping to M=8..15 simply
// re-read A rows 0..7 (row & 7): their products land only in C rows 8..15,
// which are never stored -> no predication, no exec-mask divergence.
// A layout (16x4): lanes 0-15 M=lane K={k0,k0+1}; lanes 16-31 M=lane-16 K={k0+2,k0+3}
// B layout (4x16): vgpr j: lanes 0-15 K=k0+j N=lane; lanes 16-31 K=k0+j+2
// C layout: vgpr j: lanes 0-15 -> (M=j, N=lane); lanes 16-31 -> (M=8+j, ...)
// Two independent accumulators break the WMMA->WMMA RAW chain.
// ---------------------------------------------------------------------------
__global__ void k_gemm_wmma(const float* __restrict__ A, const float* __restrict__ W,
                            float* __restrict__ out, int K, int N, int Mrows) {
  const int lane = threadIdx.x & 31;
  const int wave = threadIdx.x >> 5;
  const long n0 = (long)(blockIdx.x * (blockDim.x >> 5) + wave) * 16;
  if (n0 >= N) return;
  const int col  = lane & 15;       // N index for B/C
  const int half = lane >> 4;       // 0: K pair {0,1}, 1: K pair {2,3}
  const int rowe = lane & 7;        // A row (M=8..15 lanes alias rows 0..7)

  const float* Arow = A + (long)rowe * K + half * 2;  // 8B aligned
  const float* Wcol = W + n0 + col;

  v8f c0 = {}, c1 = {};
  int k0 = 0;
  #pragma unroll 2
  for (; k0 + 8 <= K; k0 += 8) {
    const float2 af0 = *(const float2*)(Arow + k0);
    const float2 af1 = *(const float2*)(Arow + k0 + 4);
    const float* wp0 = Wcol + (long)(k0 + half * 2) * N;
    const float* wp1 = Wcol + (long)(k0 + 4 + half * 2) * N;
    v2f a0, a1, b0, b1;
    a0.x = af0.x; a0.y = af0.y;
    a1.x = af1.x; a1.y = af1.y;
    b0.x = wp0[0]; b0.y = wp0[N];
    b1.x = wp1[0]; b1.y = wp1[N];
    c0 = __builtin_amdgcn_wmma_f32_16x16x4_f32(false, a0, false, b0, (short)0, c0,
                                               false, false);
    c1 = __builtin_amdgcn_wmma_f32_16x16x4_f32(false, a1, false, b1, (short)0, c1,
                                               false, false);
  }
  for (; k0 < K; k0 += 4) {  // tail (unused for K in {256,1024,2048,3072})
    const float2 af0 = *(const float2*)(Arow + k0);
    const float* wp0 = Wcol + (long)(k0 + half * 2) * N;
    v2f a0, b0;
    a0.x = af0.x; a0.y = af0.y;
    b0.x = wp0[0]; b0.y = wp0[N];
    c0 = __builtin_amdgcn_wmma_f32_16x16x4_f32(false, a0, false, b0, (short)0, c0,
                                               false, false);
  }
  const v8f c = c0 + c1;
  if (half == 0) {
    #pragma unroll
    for (int j = 0; j < 8; ++j)
      if (j < Mrows) out[(long)j * N + n0 + col] = c[j];
  }
}

// ---------------------------------------------------------------------------
// RMSNorm: out[b,c] = x[b,c] * rsqrt(mean(x^2)+eps) * w[c].  grid=B, block=256.
// ---------------------------------------------------------------------------
__global__ void k_rmsnorm(const float* __restrict__ x, const float* __restrict__ w,
                          float* __restrict__ out, int N) {
  const int b = blockIdx.x, t = threadIdx.x;
  __shared__ float red[256];
  float ss = 0.f;
  for (int c = t; c < N; c += 256) { float v = x[b * N + c]; ss += v * v; }
  red[t] = ss; __syncthreads();
  for (int s = 128; s > 0; s >>= 1) { if (t < s) red[t] += red[t + s]; __syncthreads(); }
  const float rms = rsqrtf(red[0] / (float)N + EPSN);
  for (int c = t; c < N; c += 256) out[b * N + c] = x[b * N + c] * rms * w[c];
}

// ---------------------------------------------------------------------------
// x = (x + rmsnorm(f, w)) * scale    (scale_p==nullptr -> 1.0)
// ---------------------------------------------------------------------------
__global__ void k_resid_norm(float* __restrict__ x, const float* __restrict__ f,
                             const float* __restrict__ w, const float* scale_p, int N) {
  const int b = blockIdx.x, t = threadIdx.x;
  __shared__ float red[256];
  float ss = 0.f;
  for (int c = t; c < N; c += 256) { float v = f[b * N + c]; ss += v * v; }
  red[t] = ss; __syncthreads();
  for (int s = 128; s > 0; s >>= 1) { if (t < s) red[t] += red[t + s]; __syncthreads(); }
  const float rms = rsqrtf(red[0] / (float)N + EPSN);
  const float sc  = scale_p ? scale_p[0] : 1.0f;
  for (int c = t; c < N; c += 256)
    x[b * N + c] = (x[b * N + c] + f[b * N + c] * rms * w[c]) * sc;
}

// ---------------------------------------------------------------------------
// Per-head RMSNorm + RoPE (partial rotary).  grid = B*H, block = D (256/512).
// q layout: (B, H*D) row-major; this block handles q[b, h*D : (h+1)*D].
// ---------------------------------------------------------------------------
__global__ void k_qnorm_rope(float* __restrict__ q, const float* __restrict__ w,
                             const int* __restrict__ pos_p, int D, int ra, float theta) {
  const int bh = blockIdx.x;
  const int c  = threadIdx.x;           // blockDim.x == D (power of 2)
  __shared__ float qn[512];
  __shared__ float red[512];
  float* base = q + (long)bh * D;
  float v = base[c];
  red[c] = v * v; __syncthreads();
  for (int s = D >> 1; s > 0; s >>= 1) { if (c < s) red[c] += red[c + s]; __syncthreads(); }
  const float rms = rsqrtf(red[0] / (float)D + EPSN);
  qn[c] = v * rms * w[c];
  __syncthreads();
  const float pos = (float)pos_p[0];
  const int hD = D >> 1;
  const int j = (c < hD) ? c : (c - hD);
  const float inv = (j < ra) ? __expf(-(2.0f * (float)j / (float)D) * __logf(theta)) : 0.0f;
  const float fr = pos * inv;
  const float cv = cosf(fr), sv = sinf(fr);
  const float rot = (c < hD) ? -qn[c + hD] : qn[c - hD];
  base[c] = qn[c] * cv + rot * sv;
}

// ---------------------------------------------------------------------------
// scores[bh, s] = dot(q[bh,:], K[bh,s,:]) + mask[b,s]
// grid = B*H*(S/64); block = 256 (8 waves, wave handles 1 s per iter x 8 iters).
// Coalesced float4 streaming of the K cache (dominant HBM traffic).
// ---------------------------------------------------------------------------
__global__ void k_scores(const float* __restrict__ q, const float* __restrict__ Kc,
                         const float* __restrict__ mask, float* __restrict__ sc,
                         int D, int S, int H) {
  const int chunks = S / 64;
  const int bh  = blockIdx.x / chunks;
  const int s0  = (blockIdx.x % chunks) * 64;
  const int b   = bh / H;
  __shared__ __align__(16) float lq[512];
  for (int c = threadIdx.x; c < D; c += blockDim.x) lq[c] = q[(long)bh * D + c];
  __syncthreads();
  const int lane = threadIdx.x & 31, wave = threadIdx.x >> 5;
  const float4* lq4 = (const float4*)lq;
  const int steps = D / 128;
  for (int i = 0; i < 8; ++i) {
    const int s = s0 + i * 8 + wave;
    const float4* kr = (const float4*)(Kc + ((long)bh * S + s) * D);
    float sum = 0.f;
    #pragma unroll
    for (int u = 0; u < 4; ++u) {
      if (u < steps) {
        float4 kk = kr[u * 32 + lane];
        float4 qq = lq4[u * 32 + lane];
        sum += kk.x * qq.x + kk.y * qq.y + kk.z * qq.z + kk.w * qq.w;
      }
    }
    for (int off = 16; off > 0; off >>= 1) sum += __shfl_xor(sum, off, 32);
    if (lane == 0) sc[(long)bh * S + s] = sum + mask[(long)b * S + s];
  }
}

// ---------------------------------------------------------------------------
// In-place softmax over S per (b,h).  grid = B*H, block = 256.
// ---------------------------------------------------------------------------
__global__ void k_softmax(float* __restrict__ sc, int S) {
  const int bh = blockIdx.x, t = threadIdx.x;
  float* row = sc + (long)bh * S;
  __shared__ float red[256];
  float m = -1e30f;
  for (int s = t; s < S; s += 256) m = fmaxf(m, row[s]);
  red[t] = m; __syncthreads();
  for (int s = 128; s > 0; s >>= 1) { if (t < s) red[t] = fmaxf(red[t], red[t + s]); __syncthreads(); }
  m = red[0]; __syncthreads();
  float sum = 0.f;
  for (int s = t; s < S; s += 256) { float e = expf(row[s] - m); row[s] = e; sum += e; }
  red[t] = sum; __syncthreads();
  for (int s = 128; s > 0; s >>= 1) { if (t < s) red[t] += red[t + s]; __syncthreads(); }
  const float inv = 1.0f / red[0];
  for (int s = t; s < S; s += 256) row[s] *= inv;
}

// ---------------------------------------------------------------------------
// out[b, h*D+d] = sum_s attn[bh,s] * V[bh,d,s]   (V stored (B,H,D,S))
// grid = B*H*(D/8); block = 256 (attn row staged in LDS, V streamed float4).
// ---------------------------------------------------------------------------
__global__ void k_av(const float* __restrict__ attn, const float* __restrict__ V,
                     float* __restrict__ out, int D, int S) {
  const int dch = D / 8;
  const int bh = blockIdx.x / dch;
  const int d0 = (blockIdx.x % dch) * 8;
  __shared__ __align__(16) float la[4096];
  const float4* ag = (const float4*)(attn + (long)bh * S);
  float4* la4 = (float4*)la;
  for (int i = threadIdx.x; i < S / 4; i += blockDim.x) la4[i] = ag[i];
  __syncthreads();
  const int lane = threadIdx.x & 31, wave = threadIdx.x >> 5;
  const int d = d0 + wave;
  const float4* vr = (const float4*)(V + ((long)bh * D + d) * S);
  float sum = 0.f;
  for (int u = 0; u < S / 128; ++u) {
    float4 vv = vr[u * 32 + lane];
    float4 aa = la4[u * 32 + lane];
    sum += vv.x * aa.x + vv.y * aa.y + vv.z * aa.z + vv.w * aa.w;
  }
  for (int off = 16; off > 0; off >>= 1) sum += __shfl_xor(sum, off, 32);
  if (lane == 0) out[(long)bh * D + d] = sum;
}

// ---------------------------------------------------------------------------
// m = gelu_tanh(g) * u
// ---------------------------------------------------------------------------
__global__ void k_gelumul(const float* __restrict__ g, const float* __restrict__ u,
                          float* __restrict__ m, int n) {
  const int i = blockIdx.x * 256 + threadIdx.x;
  if (i >= n) return;
  const float x = g[i];
  const float t = tanhf(0.7978845608028654f * (x + 0.044715f * x * x * x));
  m[i] = 0.5f * x * (1.0f + t) * u[i];
}

// ---------------------------------------------------------------------------
// Host-side orchestration
// ---------------------------------------------------------------------------
static inline void launch_gemm(const float* A, const float* W, float* out,
                               int K, int N, hipStream_t s) {
  // N is always a multiple of 64 here (256,1024,1536,2048,262144)
  k_gemm_wmma<<<dim3(N / 64), dim3(128), 0, s>>>(A, W, out, K, N, 8);
}

extern "C" void kernel_launch(void* const* d_in, const int* in_sizes, int n_in,
                              void* d_out, int out_size, void* d_ws, size_t ws_size,
                              hipStream_t stream) {
  (void)in_sizes; (void)n_in; (void)out_size; (void)ws_size;
  constexpr int B = 8, H = 4, S = 4096, HID = 256, FFN = 2048;
  constexpr int VOCAB = 262144, TGT = 1536;

  int idx = 0;
  const float* act   = (const float*)d_in[idx++];
  const float* pre_w = (const float*)d_in[idx++];
  const float *wq[4], *qnw[4], *wo[4], *ln_in[4], *ln_pa[4], *ln_pf[4], *ln_pff[4];
  const float *wg[4], *wu[4], *wd[4], *lsc[4];
  for (int L = 0; L < 4; ++L) {
    wq[L]     = (const float*)d_in[idx++];
    qnw[L]    = (const float*)d_in[idx++];
    wo[L]     = (const float*)d_in[idx++];
    ln_in[L]  = (const float*)d_in[idx++];
    ln_pa[L]  = (const float*)d_in[idx++];
    ln_pf[L]  = (const float*)d_in[idx++];
    ln_pff[L] = (const float*)d_in[idx++];
    wg[L]     = (const float*)d_in[idx++];
    wu[L]     = (const float*)d_in[idx++];
    wd[L]     = (const float*)d_in[idx++];
    lsc[L]    = (const float*)d_in[idx++];
  }
  const float* final_norm = (const float*)d_in[idx++];
  const float* post_w     = (const float*)d_in[idx++];
  const float* lm_head    = (const float*)d_in[idx++];
  const float* kv13_k     = (const float*)d_in[idx++];
  const float* kv13_v     = (const float*)d_in[idx++];
  const float* kv14_k     = (const float*)d_in[idx++];
  const float* kv14_v     = (const float*)d_in[idx++];
  const float* mask_swa   = (const float*)d_in[idx++];
  const float* mask_full  = (const float*)d_in[idx++];
  const int*   pos        = (const int*)d_in[idx++];

  float* ws = (float*)d_ws;
  float* x       = ws;                 // B*256
  float* h       = x + B * HID;        // B*256
  float* q       = h + B * HID;        // B*2048 max
  float* attnout = q + B * 2048;       // B*2048 max
  float* ao      = attnout + B * 2048; // B*256
  float* g       = ao + B * HID;       // B*2048
  float* u       = g + B * FFN;        // B*2048
  float* mm      = u + B * FFN;        // B*2048
  float* sc      = mm + B * FFN;       // B*H*S = 131072

  float* logits = (float*)d_out;             // B * VOCAB
  float* proj   = logits + (long)B * VOCAB;  // B * TGT

  // x = activations @ pre_w   (8x3072 @ 3072x256)
  launch_gemm(act, pre_w, x, 3072, HID, stream);

  for (int L = 0; L < 4; ++L) {
    const bool swa = (L < 3);
    const int  D   = swa ? 256 : 512;
    const int  HD  = H * D;
    const float* Kc   = swa ? kv13_k : kv14_k;
    const float* Vc   = swa ? kv13_v : kv14_v;
    const float* mask = swa ? mask_swa : mask_full;
    const float theta = swa ? 1.0e4f : 1.0e6f;
    const int   ra    = swa ? (D / 2) : 64;   // partial_rotary: 1.0 vs 0.25

    // h = rmsnorm(x, ln_in)
    k_rmsnorm<<<dim3(B), dim3(256), 0, stream>>>(x, ln_in[L], h, HID);
    // q = h @ wq  (8x256 @ 256xHD)
    launch_gemm(h, wq[L], q, HID, HD, stream);
    // per-head rmsnorm + RoPE (in place)
    k_qnorm_rope<<<dim3(B * H), dim3(D), 0, stream>>>(q, qnw[L], pos, D, ra, theta);
    // scores = q . K^T + mask
    k_scores<<<dim3(B * H * (S / 64)), dim3(256), 0, stream>>>(q, Kc, mask, sc, D, S, H);
    // softmax over S
    k_softmax<<<dim3(B * H), dim3(256), 0, stream>>>(sc, S);
    // attnout = attn @ V  (V is (B,H,D,S))
    k_av<<<dim3(B * H * (D / 8)), dim3(256), 0, stream>>>(sc, Vc, attnout, D, S);
    // ao = attnout @ wo
    launch_gemm(attnout, wo[L], ao, HD, HID, stream);
    // x = x + rmsnorm(ao, ln_post_attn)
    k_resid_norm<<<dim3(B), dim3(256), 0, stream>>>(x, ao, ln_pa[L], nullptr, HID);
    // h = rmsnorm(x, ln_pre_ffn)
    k_rmsnorm<<<dim3(B), dim3(256), 0, stream>>>(x, ln_pf[L], h, HID);
    // MLP
    launch_gemm(h, wg[L], g, HID, FFN, stream);
    launch_gemm(h, wu[L], u, HID, FFN, stream);
    k_gelumul<<<dim3((B * FFN) / 256), dim3(256), 0, stream>>>(g, u, mm, B * FFN);
    launch_gemm(mm, wd[L], ao, FFN, HID, stream);
    // x = (x + rmsnorm(mlp, ln_post_ffn)) * layer_scalar
    k_resid_norm<<<dim3(B), dim3(256), 0, stream>>>(x, ao, ln_pff[L], lsc[L], HID);
  }

  // h = rmsnorm(x, final_norm)
  k_rmsnorm<<<dim3(B), dim3(256), 0, stream>>>(x, final_norm, h, HID);
  // logits = h @ lm_head  (the 268 MB streaming GEMM)
  launch_gemm(h, lm_head, logits, HID, VOCAB, stream);
  // proj_act = h @ post_w
  launch_gemm(h, post_w, proj, HID, TGT, stream);
}